// PartCEMClip_59682865545398
// MI455X (gfx1250) — compile-verified
//
#include <hip/hip_runtime.h>
#include <cstdint>
#include <cstddef>

// PartCEM fused kernel for MI455X (gfx1250, wave32, WMMA f32 16x16x4).
// B=128, C=1024, H=W=24 (HW=576), K=9 (padded to 16), NUM_CLASSES=200.

typedef __attribute__((ext_vector_type(2))) float v2f;
typedef __attribute__((ext_vector_type(8))) float v8f;

#define WMMA_F32(a, b, c) \
  __builtin_amdgcn_wmma_f32_16x16x4_f32(false, (a), false, (b), (short)0, (c), false, false)

static constexpr int CC  = 1024;
static constexpr int HW  = 576;
static constexpr int KP  = 9;     // real prototypes
static constexpr int NCL = 200;

// ---------------------------------------------------------------------------
// Kernel 0: zero-pad prototypes to 16x1024 in ws, compute p2[m]=||proto_m||^2
// ---------------------------------------------------------------------------
__global__ void k_prep(const float* __restrict__ proto,
                       float* __restrict__ wsproto,   // 16*1024 floats
                       float* __restrict__ wsp2) {    // 16 floats
  __shared__ float red[256];
  const int tid = threadIdx.x;
  for (int idx = tid; idx < 16 * CC; idx += 256) {
    const int r = idx >> 10, c = idx & (CC - 1);
    wsproto[idx] = (r < KP) ? proto[r * CC + c] : 0.0f;
  }
  // p2: 16 threads per row m (rows >= 9 -> 0)
  const int m = tid >> 4, j = tid & 15;
  float s = 0.0f;
  if (m < KP) {
    const int c0 = j * 64;
    for (int c = c0; c < c0 + 64; ++c) {
      const float v = proto[m * CC + c];
      s += v * v;
    }
  }
  red[tid] = s;
  __syncthreads();
  if (j == 0) {
    float t = 0.0f;
    for (int q = 0; q < 16; ++q) t += red[m * 16 + q];
    wsp2[m] = (m < KP) ? t : 0.0f;
  }
}

// ---------------------------------------------------------------------------
// Kernel 1: per-batch fused  (G = proto x_b) -> softmax maps -> parts
// grid = 128 blocks of 288 threads (9 waves).
// ---------------------------------------------------------------------------
__global__ void __launch_bounds__(288)
k_main(const float* __restrict__ x,
       const float* __restrict__ wsproto,
       const float* __restrict__ wsp2,
       float* __restrict__ out_parts,   // (128,9,1024)
       float* __restrict__ out_maps) {  // (128,9,576)
  __shared__ float mapsLDS[16 * 578];   // [k16][hw] padded stride (even, !=0 mod 64)

  const int b    = blockIdx.x;
  const int tid  = threadIdx.x;
  const int lane = tid & 31;
  const int wave = tid >> 5;            // 0..8
  const int hi   = lane >> 4;           // 0 for lanes 0-15, 1 for 16-31
  const int n    = lane & 15;

  const float* __restrict__ xb = x + (size_t)b * CC * HW;

  // ================= Phase A: G[k16][hw] = proto_pad @ x_b ================
  // Each wave owns 4 hw-tiles of 16 columns: hwbase .. hwbase+63.
  const int hwbase = wave * 64;

  v8f a0 = {}, a1 = {}, a2 = {}, a3 = {};
  float s20 = 0.f, s21 = 0.f, s22 = 0.f, s23 = 0.f;

  // A operand: proto[m= lane&15][c + 2*hi .. +1]  (b64, 8B aligned)
  const float* __restrict__ pA = wsproto + (n * CC + 2 * hi);

  for (int c = 0; c < CC; c += 4) {
    const v2f a = *(const v2f*)(pA + c);
    // B operand rows: x[c + 2*hi][hw0+n] (v.x) and x[c + 2*hi + 1][hw0+n] (v.y)
    const float* __restrict__ pX = xb + (size_t)(c + 2 * hi) * HW + n;
    v2f b0 = { pX[hwbase +  0], pX[HW + hwbase +  0] };
    v2f b1 = { pX[hwbase + 16], pX[HW + hwbase + 16] };
    v2f b2 = { pX[hwbase + 32], pX[HW + hwbase + 32] };
    v2f b3 = { pX[hwbase + 48], pX[HW + hwbase + 48] };
    s20 += b0.x * b0.x + b0.y * b0.y;
    s21 += b1.x * b1.x + b1.y * b1.y;
    s22 += b2.x * b2.x + b2.y * b2.y;
    s23 += b3.x * b3.x + b3.y * b3.y;
    a0 = WMMA_F32(a, b0, a0);
    a1 = WMMA_F32(a, b1, a1);
    a2 = WMMA_F32(a, b2, a2);
    a3 = WMMA_F32(a, b3, a3);
  }

  // ||x||^2 per column: low half held c,c+1 terms; high half c+2,c+3.
  const float x20 = s20 + __shfl_xor(s20, 16);
  const float x21 = s21 + __shfl_xor(s21, 16);
  const float x22 = s22 + __shfl_xor(s22, 16);
  const float x23 = s23 + __shfl_xor(s23, 16);

  float p2v[8];
#pragma unroll
  for (int r = 0; r < 8; ++r) p2v[r] = wsp2[8 * hi + r];

  auto do_tile = [&](v8f A, float x2t, int hw0) {
    float dist[8];
#pragma unroll
    for (int r = 0; r < 8; ++r) {
      float d2 = x2t + p2v[r] - 2.0f * A[r];
      d2 = fmaxf(d2, 1e-12f);
      dist[r] = sqrtf(d2);
    }
    // m=8 lives in the high half (row r=0 of lanes 16..31)
    const float d8 = __shfl_down(dist[0], 16);
    float mn = dist[0];
#pragma unroll
    for (int r = 1; r < 8; ++r) mn = fminf(mn, dist[r]);
    mn = fminf(mn, d8);
    float e[8], sum = 0.0f;
#pragma unroll
    for (int r = 0; r < 8; ++r) { e[r] = __expf(mn - dist[r]); sum += e[r]; }
    const float e8 = __expf(mn - d8);
    sum += e8;
    const float inv = 1.0f / sum;
    const int hw = hw0 + n;
    if (lane < 16) {
#pragma unroll
      for (int m = 0; m < 8; ++m) {
        const float v = e[m] * inv;
        out_maps[((size_t)b * KP + m) * HW + hw] = v;
        mapsLDS[m * 578 + hw] = v;
      }
      const float v8_ = e8 * inv;
      out_maps[((size_t)b * KP + 8) * HW + hw] = v8_;
      mapsLDS[8 * 578 + hw] = v8_;
    } else {
#pragma unroll
      for (int m = KP; m < 16; ++m) mapsLDS[m * 578 + hw] = 0.0f;
    }
  };
  do_tile(a0, x20, hwbase);
  do_tile(a1, x21, hwbase + 16);
  do_tile(a2, x22, hwbase + 32);
  do_tile(a3, x23, hwbase + 48);

  __syncthreads();

  // ============ Phase B: parts^T[c][k] = x_b @ maps^T  (WMMA) ============
  const float inv_hw = 1.0f / (float)HW;
  for (int ct = wave; ct < CC / 16; ct += 9) {
    const int c0 = ct * 16;
    v8f acc = {};
    // A: x[c0 + (lane&15)][hw + 2*hi .. +1]  (b64, 8B aligned)
    const float* __restrict__ pAx = xb + (size_t)(c0 + n) * HW + 2 * hi;
    // B: maps[lane&15][hw + 2*hi .. +1] from LDS (b64, 8B aligned)
    const float* __restrict__ pB = &mapsLDS[n * 578 + 2 * hi];
    for (int hw = 0; hw < HW; hw += 4) {
      const v2f a  = *(const v2f*)(pAx + hw);
      const v2f bb = *(const v2f*)(pB + hw);
      acc = WMMA_F32(a, bb, acc);
    }
    const int k = n;
    if (k < KP) {
#pragma unroll
      for (int r = 0; r < 8; ++r) {
        const int c = c0 + r + 8 * hi;
        out_parts[((size_t)b * KP + k) * CC + c] = acc[r] * inv_hw;
      }
    }
  }
}

// ---------------------------------------------------------------------------
// Kernel 2: logits^T = fc_w @ (parts*mod)^T + fc_b   (WMMA)
// grid = 128 blocks of 416 threads (13 waves, one 16-class tile each).
// ---------------------------------------------------------------------------
__global__ void __launch_bounds__(416)
k_logits(const float* __restrict__ parts,
         const float* __restrict__ mod,     // (1,9,1024)
         const float* __restrict__ fcw,     // (200,1024)
         const float* __restrict__ fcb,     // (200)
         float* __restrict__ out_logits) {  // (128,9,200)
  __shared__ float pLDS[16 * CC];           // [c][k16] = parts*mod, rows k>=9 zero
  const int b   = blockIdx.x;
  const int tid = threadIdx.x;
  for (int idx = tid; idx < 16 * CC; idx += 416) {
    const int c = idx >> 4, k = idx & 15;
    float v = 0.0f;
    if (k < KP) v = parts[((size_t)b * KP + k) * CC + c] * mod[k * CC + c];
    pLDS[idx] = v;
  }
  __syncthreads();

  const int lane = tid & 31;
  const int wave = tid >> 5;   // 0..12
  const int hi   = lane >> 4;
  const int n    = lane & 15;
  const int n0   = wave * 16;

  int row = n0 + n;            // class row for A operand; clamp partial tile
  if (row > NCL - 1) row = NCL - 1;
  const float* __restrict__ pA = fcw + (size_t)row * CC + 2 * hi;

  v8f acc = {};
  for (int c = 0; c < CC; c += 4) {
    const v2f a = *(const v2f*)(pA + c);
    const int cB = c + 2 * hi;
    v2f bb = { pLDS[cB * 16 + n], pLDS[(cB + 1) * 16 + n] };
    acc = WMMA_F32(a, bb, acc);
  }

  const int k = n;
  if (k < KP) {
#pragma unroll
    for (int r = 0; r < 8; ++r) {
      const int cls = n0 + r + 8 * hi;
      if (cls < NCL)
        out_logits[((size_t)b * KP + k) * NCL + cls] = acc[r] + fcb[cls];
    }
  }
}

// ---------------------------------------------------------------------------
extern "C" void kernel_launch(void* const* d_in, const int* in_sizes, int n_in,
                              void* d_out, int out_size, void* d_ws, size_t ws_size,
                              hipStream_t stream) {
  (void)in_sizes; (void)n_in; (void)out_size; (void)ws_size;
  const float* x     = (const float*)d_in[0];
  const float* proto = (const float*)d_in[1];
  const float* mod   = (const float*)d_in[2];
  const float* fcw   = (const float*)d_in[3];
  const float* fcb   = (const float*)d_in[4];

  float* out    = (float*)d_out;
  float* parts  = out;                                  // 128*9*1024
  float* maps   = parts + (size_t)128 * KP * CC;        // 128*9*576
  float* logits = maps  + (size_t)128 * KP * HW;        // 128*9*200

  float* wsproto = (float*)d_ws;            // 16*1024 floats
  float* wsp2    = wsproto + 16 * CC;       // 16 floats

  k_prep  <<<1,   256, 0, stream>>>(proto, wsproto, wsp2);
  k_main  <<<128, 288, 0, stream>>>(x, wsproto, wsp2, parts, maps);
  k_logits<<<128, 416, 0, stream>>>(parts, mod, fcw, fcb, logits);
}